// GraphAutoEncoder_26645977104907
// MI455X (gfx1250) — compile-verified
//
#include <hip/hip_runtime.h>

// Problem constants (match reference)
#define NN 100000      // nodes
#define NE 1250000     // edges
// dims: L1 64->64, L2 64->32, L3 32->64, L4 64->64

typedef __attribute__((ext_vector_type(2))) float v2f;
typedef __attribute__((ext_vector_type(8))) float v8f;

// ---------------------------------------------------------------------------
// Degree / normalization: dis[i] = rsqrt(1 + #edges with dst==i)
// ---------------------------------------------------------------------------
__global__ void k_fill_ones(float* __restrict__ p, int n) {
    int i = blockIdx.x * blockDim.x + threadIdx.x;
    if (i < n) p[i] = 1.0f;   // self-loop contributes 1 to every degree
}

__global__ void k_degree_count(const int* __restrict__ dst, float* __restrict__ deg, int e) {
    int i = blockIdx.x * blockDim.x + threadIdx.x;
    if (i < e) unsafeAtomicAdd(&deg[dst[i]], 1.0f);
}

__global__ void k_rsqrt(float* __restrict__ p, int n) {
    int i = blockIdx.x * blockDim.x + threadIdx.x;
    if (i < n) p[i] = rsqrtf(p[i]);   // deg >= 1 always (self loop)
}

// ---------------------------------------------------------------------------
// GEMM: hw = h @ W  via V_WMMA_F32_16X16X4_F32.
// One wave computes a full 16 x DOUT strip: the A fragment for each K-quad is
// loaded once and reused across all DOUT/16 column tiles (4x fewer h reads
// than one-tile-per-wave). Also seeds agg with the self-loop term
// agg[i,:] = hw[i,:] * dis[i]^2 so no separate zero-fill pass is needed.
// h: [NN, DIN] row-major f32;  W: [DIN, DOUT] row-major f32.
// ---------------------------------------------------------------------------
template <int DIN, int DOUT>
__global__ void k_gcn_gemm(const float* __restrict__ h, const float* __restrict__ W,
                           const float* __restrict__ dis,
                           float* __restrict__ hw, float* __restrict__ agg) {
    constexpr int CT = DOUT / 16;            // column tiles per wave
    const int lane = threadIdx.x & 31;
    const int wave = (blockIdx.x * blockDim.x + threadIdx.x) >> 5;
    if (wave >= NN / 16) return;             // NN % 16 == 0

    const int row0 = wave * 16;              // node block
    const int half = lane >> 4;              // 0: lanes 0-15, 1: lanes 16-31
    const int ml   = lane & 15;

    v8f acc[CT];
#pragma unroll
    for (int c = 0; c < CT; ++c) acc[c] = (v8f){};

#pragma unroll
    for (int k0 = 0; k0 < DIN; k0 += 4) {
        const int ka = k0 + 2 * half;
        // A fragment: M = ml (per half), K = {ka, ka+1} in vgpr {0,1}
        const float* hp = h + (size_t)(row0 + ml) * DIN + ka;
        v2f a; a.x = hp[0]; a.y = hp[1];
        const float* wp0 = W + (size_t)ka * DOUT + ml;
        const float* wp1 = wp0 + DOUT;
#pragma unroll
        for (int c = 0; c < CT; ++c) {
            // B fragment: N = ml (per half), K = {ka, ka+1} in vgpr {0,1}
            v2f b; b.x = wp0[c * 16]; b.y = wp1[c * 16];
            acc[c] = __builtin_amdgcn_wmma_f32_16x16x4_f32(
                /*neg_a=*/false, a, /*neg_b=*/false, b,
                /*c_mod=*/(short)0, acc[c], /*reuse_a=*/false, /*reuse_b=*/false);
        }
    }

    // Self-loop factors dis[m]^2 hoisted out of the column-tile store loop
    float dsq[8];
#pragma unroll
    for (int r = 0; r < 8; ++r) {
        const float d = dis[row0 + r + 8 * half];
        dsq[r] = d * d;
    }

    // D layout: vgpr r -> M = r + 8*half, N = ml (per column tile)
#pragma unroll
    for (int c = 0; c < CT; ++c) {
#pragma unroll
        for (int r = 0; r < 8; ++r) {
            const int m = row0 + r + 8 * half;
            const size_t idx = (size_t)m * DOUT + c * 16 + ml;
            const float v = acc[c][r];
            hw[idx]  = v;
            agg[idx] = v * dsq[r];
        }
    }
}

// ---------------------------------------------------------------------------
// Edge scatter: agg[dst,:] += hw[src,:] * dis[src]*dis[dst]
// One thread per (edge, feature-pair); consecutive threads cover consecutive
// features of the same edge -> coalesced hw reads; f32 HW atomics to L2.
// ---------------------------------------------------------------------------
template <int DIM>
__global__ void k_gcn_scatter(const float* __restrict__ hw,
                              const int* __restrict__ src, const int* __restrict__ dst,
                              const float* __restrict__ dis, float* __restrict__ agg) {
    const int PAIRS = DIM / 2;
    long long t = (long long)blockIdx.x * blockDim.x + threadIdx.x;
    const long long total = (long long)NE * PAIRS;
    if (t >= total) return;
    const int e = (int)(t / PAIRS);
    const int f = (int)(t % PAIRS) * 2;
    const int s = src[e];
    const int d = dst[e];
    const float w = dis[s] * dis[d];
    const float* hp = hw + (size_t)s * DIM + f;
    float* ap = agg + (size_t)d * DIM + f;
    unsafeAtomicAdd(ap,     hp[0] * w);
    unsafeAtomicAdd(ap + 1, hp[1] * w);
}

// ---------------------------------------------------------------------------
// out = (agg + b) with optional ReLU
// ---------------------------------------------------------------------------
template <int DIM, bool RELU>
__global__ void k_bias_act(const float* __restrict__ agg, const float* __restrict__ b,
                           float* __restrict__ out) {
    long long t = (long long)blockIdx.x * blockDim.x + threadIdx.x;
    const long long total = (long long)NN * DIM;
    if (t >= total) return;
    const int f = (int)(t % DIM);
    float v = agg[t] + b[f];
    out[t] = RELU ? fmaxf(v, 0.0f) : v;
}

// ---------------------------------------------------------------------------
// Launch
// ---------------------------------------------------------------------------
extern "C" void kernel_launch(void* const* d_in, const int* in_sizes, int n_in,
                              void* d_out, int out_size, void* d_ws, size_t ws_size,
                              hipStream_t stream) {
    (void)in_sizes; (void)n_in; (void)out_size; (void)ws_size;

    const float* x  = (const float*)d_in[0];
    const int*   ei = (const int*)d_in[1];        // edge_index [2, NE]
    const int*   src = ei;
    const int*   dst = ei + NE;
    const float* W1 = (const float*)d_in[2]; const float* b1 = (const float*)d_in[3];
    const float* W2 = (const float*)d_in[4]; const float* b2 = (const float*)d_in[5];
    const float* W3 = (const float*)d_in[6]; const float* b3 = (const float*)d_in[7];
    const float* W4 = (const float*)d_in[8]; const float* b4 = (const float*)d_in[9];

    float* xrec = (float*)d_out;              // [NN, 64]
    float* z    = xrec + (size_t)NN * 64;     // [NN, 32]

    float* ws   = (float*)d_ws;
    float* dis  = ws;                         // NN floats (deg -> rsqrt in place)
    float* hw   = ws + 102400;                // NN*64
    float* agg  = hw + (size_t)NN * 64;       // NN*64
    float* hbuf = agg + (size_t)NN * 64;      // NN*64 (h1, then d1)

    const int B = 256;
    const int WAVES_PER_BLOCK = B / 32;

    // --- normalization coefficients ---
    k_fill_ones<<<(NN + B - 1) / B, B, 0, stream>>>(dis, NN);
    k_degree_count<<<(NE + B - 1) / B, B, 0, stream>>>(dst, dis, NE);
    k_rsqrt<<<(NN + B - 1) / B, B, 0, stream>>>(dis, NN);

    // One wave per 16-row strip (all column tiles fused into the wave)
    const int strips = NN / 16;   // 6250
    const int gGemm  = (strips + WAVES_PER_BLOCK - 1) / WAVES_PER_BLOCK;

    const long long sc64 = (long long)NE * 32;  // edge*featpair threads, dim 64
    const long long sc32 = (long long)NE * 16;  // dim 32
    const int gSc64 = (int)((sc64 + B - 1) / B);
    const int gSc32 = (int)((sc32 + B - 1) / B);
    const int gEw64 = (int)(((long long)NN * 64 + B - 1) / B);
    const int gEw32 = (int)(((long long)NN * 32 + B - 1) / B);

    // --- Layer 1: 64 -> 64, ReLU -> hbuf (h1) ---
    k_gcn_gemm<64, 64><<<gGemm, B, 0, stream>>>(x, W1, dis, hw, agg);
    k_gcn_scatter<64><<<gSc64, B, 0, stream>>>(hw, src, dst, dis, agg);
    k_bias_act<64, true><<<gEw64, B, 0, stream>>>(agg, b1, hbuf);

    // --- Layer 2: 64 -> 32, ReLU -> z (in d_out) ---
    k_gcn_gemm<64, 32><<<gGemm, B, 0, stream>>>(hbuf, W2, dis, hw, agg);
    k_gcn_scatter<32><<<gSc32, B, 0, stream>>>(hw, src, dst, dis, agg);
    k_bias_act<32, true><<<gEw32, B, 0, stream>>>(agg, b2, z);

    // --- Layer 3: 32 -> 64, ReLU -> hbuf (d1) ---
    k_gcn_gemm<32, 64><<<gGemm, B, 0, stream>>>(z, W3, dis, hw, agg);
    k_gcn_scatter<64><<<gSc64, B, 0, stream>>>(hw, src, dst, dis, agg);
    k_bias_act<64, true><<<gEw64, B, 0, stream>>>(agg, b3, hbuf);

    // --- Layer 4: 64 -> 64, no ReLU -> x_recon (in d_out) ---
    k_gcn_gemm<64, 64><<<gGemm, B, 0, stream>>>(hbuf, W4, dis, hw, agg);
    k_gcn_scatter<64><<<gSc64, B, 0, stream>>>(hw, src, dst, dis, agg);
    k_bias_act<64, false><<<gEw64, B, 0, stream>>>(agg, b4, xrec);
}